// hypergraph_part_40218073760224
// MI455X (gfx1250) — compile-verified
//
#include <hip/hip_runtime.h>
#include <hip/hip_bf16.h>

typedef __attribute__((ext_vector_type(2))) float v2f;
typedef __attribute__((ext_vector_type(8))) float v8f;

#define D_N   512
#define M_N   1024
#define F_N   64
#define NM    1536          // D+M

// ---------------- workspace layout (floats) ----------------
#define OFF_XW     0                         // [1536*64]
#define OFF_S      (OFF_XW   + NM*F_N)       // [1536]
#define OFF_T      (OFF_S    + NM)           // [512]
#define OFF_EFEAT  (OFF_T    + D_N)          // [512*64]
#define OFF_A0     (OFF_EFEAT+ D_N*F_N)      // [512]
#define OFF_PE0S   (OFF_A0   + D_N)          // [64]
#define OFF_PE1S   (OFF_PE0S + F_N)          // [64]
#define OFF_XCS    (OFF_PE1S + F_N)          // [64]
#define OFF_XMS    (OFF_XCS  + F_N)          // [64]
#define OFF_VATT   (OFF_XMS  + F_N)          // [64]
#define OFF_WATT   (OFF_VATT + F_N)          // [64]
#define OFF_DIA    (OFF_WATT + F_N)          // [64]
#define OFF_MED    (OFF_DIA  + F_N)          // [64]

// ---------------------------------------------------------------------------
// Kernel 1: gathered row-sums: pe0[c_it].sum, pe1[m_it].sum, x_c.sum, x_m.sum
// 1 block, 256 threads (f = tid&63 feature lane, g = tid>>6 row group)
// ---------------------------------------------------------------------------
__global__ void k_input_sums(const int* __restrict__ c_it, const int* __restrict__ m_it,
                             const float* __restrict__ c_emb, const float* __restrict__ m_emb,
                             const float* __restrict__ pe0, const float* __restrict__ pe1,
                             float* __restrict__ PE0S, float* __restrict__ PE1S,
                             float* __restrict__ XCS,  float* __restrict__ XMS) {
    __shared__ float red[256];
    int tid = threadIdx.x, f = tid & 63, g = tid >> 6;
    float a_pe0 = 0.f, a_xc = 0.f, a_pe1 = 0.f, a_xm = 0.f;
    for (int i = g; i < D_N; i += 4) {
        int idx = c_it[i];
        a_pe0 += pe0[(size_t)idx * F_N + f];
        a_xc  += c_emb[(size_t)idx * F_N + f];
    }
    for (int j = g; j < M_N; j += 4) {
        int idx = m_it[j];
        a_pe1 += pe1[(size_t)idx * F_N + f];
        a_xm  += m_emb[(size_t)idx * F_N + f];
    }
    red[tid] = a_pe0; __syncthreads();
    if (tid < 64) PE0S[tid] = red[tid] + red[tid+64] + red[tid+128] + red[tid+192];
    __syncthreads();
    red[tid] = a_pe1; __syncthreads();
    if (tid < 64) PE1S[tid] = red[tid] + red[tid+64] + red[tid+128] + red[tid+192];
    __syncthreads();
    red[tid] = a_xc; __syncthreads();
    if (tid < 64) XCS[tid] = red[tid] + red[tid+64] + red[tid+128] + red[tid+192];
    __syncthreads();
    red[tid] = a_xm; __syncthreads();
    if (tid < 64) XMS[tid] = red[tid] + red[tid+64] + red[tid+128] + red[tid+192];
}

// ---------------------------------------------------------------------------
// Kernel 2: tiny vectors: v_att = Wᵀ att[:64], w_att = Wᵀ att[64:],
// dia_vec = conv_b + (xc_sum/D)@conv_wᵀ, med_vec likewise. 1 block, 64 thr.
// ---------------------------------------------------------------------------
__global__ void k_prep(const float* __restrict__ gat_w, const float* __restrict__ gat_att,
                       const float* __restrict__ conv_w, const float* __restrict__ conv_b,
                       const float* __restrict__ XCS, const float* __restrict__ XMS,
                       float* __restrict__ VATT, float* __restrict__ WATT,
                       float* __restrict__ DIA,  float* __restrict__ MED) {
    int k = threadIdx.x;   // 0..63
    float va = 0.f, wa = 0.f;
    for (int f = 0; f < F_N; ++f) {
        float wv = gat_w[(size_t)f * F_N + k];
        va += wv * gat_att[f];
        wa += wv * gat_att[F_N + f];
    }
    VATT[k] = va; WATT[k] = wa;
    float ad = 0.f, am = 0.f;
    for (int kk = 0; kk < F_N; ++kk) {
        float cw = conv_w[(size_t)k * F_N + kk];   // row c = k
        ad += XCS[kk] * cw;
        am += XMS[kk] * cw;
    }
    DIA[k] = conv_b[k] + ad * (1.0f / D_N);
    MED[k] = conv_b[k] + am * (1.0f / M_N);
}

// ---------------------------------------------------------------------------
// Kernel 3: xw = comb @ gat_wᵀ  via V_WMMA_F32_16X16X4_F32.
// grid = 96*4 tiles, 32 threads (one wave) per tile, K-loop of 16 WMMAs.
// A layout (16x4 f32): lanes 0-15 hold K=k0..k0+1, lanes 16-31 K=k0+2..k0+3.
// B layout (4x16 f32): mirrored (row K striped across lanes).
// C/D layout (16x16 f32): VGPR p -> row p (lanes 0-15) / row p+8 (lanes 16-31).
// ---------------------------------------------------------------------------
__global__ void k_gemm(const int* __restrict__ c_it, const int* __restrict__ m_it,
                       const float* __restrict__ c_emb, const float* __restrict__ m_emb,
                       const float* __restrict__ gat_w, float* __restrict__ XW) {
    int tile = blockIdx.x;          // 0..383
    int tm = tile >> 2, tn = tile & 3;
    int lane = threadIdx.x;         // 0..31, wave32
    int lm = lane & 15, half = lane >> 4;
    int m = tm * 16 + lm;           // output row 0..1535
    const float* rowp = (m < D_N) ? (c_emb + (size_t)c_it[m] * F_N)
                                  : (m_emb + (size_t)m_it[m - D_N] * F_N);
    int n = tn * 16 + lm;           // output col 0..63
    const float* wrow = gat_w + (size_t)n * F_N;   // gat_wᵀ[k][n] = gat_w[n][k]

    v8f c = {};
    for (int k0 = 0; k0 < F_N; k0 += 4) {
        int ka = k0 + 2 * half;
        v2f a, b;
        a[0] = rowp[ka]; a[1] = rowp[ka + 1];
        b[0] = wrow[ka]; b[1] = wrow[ka + 1];
        c = __builtin_amdgcn_wmma_f32_16x16x4_f32(
                /*neg_a=*/false, a, /*neg_b=*/false, b,
                /*c_mod=*/(short)0, c, /*reuse_a=*/false, /*reuse_b=*/false);
    }
    int col = tn * 16 + lm;
    #pragma unroll
    for (int p = 0; p < 8; ++p) {
        int row = tm * 16 + p + 8 * half;
        XW[(size_t)row * F_N + col] = c[p];
    }
}

// ---------------------------------------------------------------------------
// Kernel 4: s[n] = comb[n]·v_att (n<1536), t[e] = hattr[e]·w_att.
// One wave per row; 8 rows per 256-thread block; grid = 2048/8 = 256.
// ---------------------------------------------------------------------------
__global__ void k_dots(const int* __restrict__ c_it, const int* __restrict__ m_it,
                       const float* __restrict__ c_emb, const float* __restrict__ m_emb,
                       const float* __restrict__ hattr,
                       const float* __restrict__ VATT, const float* __restrict__ WATT,
                       float* __restrict__ S, float* __restrict__ T) {
    int warp = threadIdx.x >> 5, lane = threadIdx.x & 31;
    int r = blockIdx.x * 8 + warp;   // 0..2047
    const float* row; const float* vec;
    if (r < D_N)      { row = c_emb + (size_t)c_it[r] * F_N;        vec = VATT; }
    else if (r < NM)  { row = m_emb + (size_t)m_it[r - D_N] * F_N;  vec = VATT; }
    else              { row = hattr + (size_t)(r - NM) * F_N;       vec = WATT; }
    float p = row[lane] * vec[lane] + row[lane + 32] * vec[lane + 32];
    for (int off = 16; off > 0; off >>= 1) p += __shfl_xor(p, off, 32);
    if (lane == 0) { if (r < NM) S[r] = p; else T[r - NM] = p; }
}

// ---------------------------------------------------------------------------
// Kernel 5: per-edge attention softmax + weighted med sum.
// 512 blocks x 256 threads; w[j] in LDS; deterministic (no atomics).
// efeat[e][f] = (w0*xw_d[e][f] + Σ_j w_j*xw_m[j][f]) / (Z * (M+1));  A0[e]=w0/Z
// ---------------------------------------------------------------------------
__global__ void k_edge(const float* __restrict__ XW, const float* __restrict__ S,
                       const float* __restrict__ T,
                       float* __restrict__ EFEAT, float* __restrict__ A0) {
    __shared__ float w[M_N];
    __shared__ float red[256];
    __shared__ float sh_mx, sh_z, sh_w0;
    int e = blockIdx.x, tid = threadIdx.x;
    float t_e = T[e];
    float b0 = S[e] + t_e; b0 = (b0 > 0.f) ? b0 : 0.2f * b0;   // leaky_relu(0.2)
    float lmax = -3.4e38f;
    #pragma unroll
    for (int q = 0; q < 4; ++q) {
        int j = tid + q * 256;
        float b = S[D_N + j] + t_e;
        b = (b > 0.f) ? b : 0.2f * b;
        w[j] = b;
        lmax = fmaxf(lmax, b);
    }
    red[tid] = lmax; __syncthreads();
    for (int s = 128; s > 0; s >>= 1) {
        if (tid < s) red[tid] = fmaxf(red[tid], red[tid + s]);
        __syncthreads();
    }
    if (tid == 0) sh_mx = fmaxf(red[0], b0);
    __syncthreads();
    float mx = sh_mx;
    float lsum = 0.f;
    #pragma unroll
    for (int q = 0; q < 4; ++q) {
        int j = tid + q * 256;
        float v = __expf(w[j] - mx);
        w[j] = v;
        lsum += v;
    }
    __syncthreads();
    red[tid] = lsum; __syncthreads();
    for (int s = 128; s > 0; s >>= 1) {
        if (tid < s) red[tid] += red[tid + s];
        __syncthreads();
    }
    if (tid == 0) { float w0 = __expf(b0 - mx); sh_w0 = w0; sh_z = red[0] + w0; }
    __syncthreads();
    float Z = sh_z, w0 = sh_w0;
    int f = tid & 63, g = tid >> 6;
    float acc = 0.f;
    for (int j = g; j < M_N; j += 4)
        acc += w[j] * XW[(size_t)(D_N + j) * F_N + f];
    __syncthreads();
    red[tid] = acc; __syncthreads();
    if (tid < 64) {
        float tot = red[tid] + red[tid + 64] + red[tid + 128] + red[tid + 192];
        float xd  = XW[(size_t)e * F_N + tid];
        float ef  = (w0 * xd + tot) / (Z * (float)(M_N + 1));
        EFEAT[(size_t)e * F_N + tid] = ef;
        if (tid == 0) A0[e] = w0 / Z;
    }
}

// ---------------------------------------------------------------------------
// Kernel 6: edge reduction + final linear + output. 1 block, 256 threads.
// Sd[c] = Σ_e a0*efeat + D*gat_b ; Sm[c] = (Σ_e (1-a0)*efeat)/D + M*gat_b
// i1[o] = pe0_sum[o] + Σ_c Sd[c]*lin_w[o][c] + Σ_c D*dia[c]*lin_w[o][64+c]
// i2[o] = pe1_sum[o] + Σ_c Sm[c]*lin_w[o][c] + Σ_c M*med[c]*lin_w[o][64+c]
// ---------------------------------------------------------------------------
__global__ void k_final(const float* __restrict__ EFEAT, const float* __restrict__ A0,
                        const float* __restrict__ gat_b,
                        const float* __restrict__ PE0S, const float* __restrict__ PE1S,
                        const float* __restrict__ DIA, const float* __restrict__ MED,
                        const float* __restrict__ lin_w, float* __restrict__ out) {
    __shared__ float red[256];
    __shared__ float sd[64], smv[64];
    int tid = threadIdx.x, f = tid & 63, g = tid >> 6;
    float aSd = 0.f, aSm = 0.f;
    for (int e = g; e < D_N; e += 4) {
        float ef = EFEAT[(size_t)e * F_N + f];
        float a0 = A0[e];
        aSd += a0 * ef;
        aSm += (1.0f - a0) * ef;
    }
    red[tid] = aSd; __syncthreads();
    if (tid < 64) sd[tid] = red[tid] + red[tid+64] + red[tid+128] + red[tid+192]
                            + (float)D_N * gat_b[tid];
    __syncthreads();
    red[tid] = aSm; __syncthreads();
    if (tid < 64) smv[tid] = (red[tid] + red[tid+64] + red[tid+128] + red[tid+192])
                             * (1.0f / D_N) + (float)M_N * gat_b[tid];
    __syncthreads();
    if (tid < 64) {
        int o = tid;
        float acc1 = PE0S[o], acc2 = PE1S[o];
        for (int c = 0; c < F_N; ++c) {
            float l1 = lin_w[(size_t)o * 128 + c];
            float l2 = lin_w[(size_t)o * 128 + 64 + c];
            acc1 += sd[c]  * l1 + (float)D_N * DIA[c] * l2;
            acc2 += smv[c] * l1 + (float)M_N * MED[c] * l2;
        }
        out[o]      = acc1;   // i1
        out[64 + o] = acc2;   // i2
    }
}

// ---------------------------------------------------------------------------
extern "C" void kernel_launch(void* const* d_in, const int* in_sizes, int n_in,
                              void* d_out, int out_size, void* d_ws, size_t ws_size,
                              hipStream_t stream) {
    const int*   c_it   = (const int*)  d_in[0];
    const int*   m_it   = (const int*)  d_in[1];
    const float* c_emb  = (const float*)d_in[2];
    const float* m_emb  = (const float*)d_in[3];
    const float* pe0    = (const float*)d_in[4];
    const float* pe1    = (const float*)d_in[5];
    const float* conv_w = (const float*)d_in[6];
    const float* conv_b = (const float*)d_in[7];
    const float* gat_w  = (const float*)d_in[8];
    const float* gat_b  = (const float*)d_in[9];
    const float* gatatt = (const float*)d_in[10];
    const float* hattr  = (const float*)d_in[11];
    const float* lin_w  = (const float*)d_in[12];
    float* out = (float*)d_out;

    float* ws    = (float*)d_ws;
    float* XW    = ws + OFF_XW;
    float* S     = ws + OFF_S;
    float* T     = ws + OFF_T;
    float* EFEAT = ws + OFF_EFEAT;
    float* A0    = ws + OFF_A0;
    float* PE0S  = ws + OFF_PE0S;
    float* PE1S  = ws + OFF_PE1S;
    float* XCS   = ws + OFF_XCS;
    float* XMS   = ws + OFF_XMS;
    float* VATT  = ws + OFF_VATT;
    float* WATT  = ws + OFF_WATT;
    float* DIA   = ws + OFF_DIA;
    float* MED   = ws + OFF_MED;

    k_input_sums<<<1, 256, 0, stream>>>(c_it, m_it, c_emb, m_emb, pe0, pe1,
                                        PE0S, PE1S, XCS, XMS);
    k_prep<<<1, 64, 0, stream>>>(gat_w, gatatt, conv_w, conv_b, XCS, XMS,
                                 VATT, WATT, DIA, MED);
    k_gemm<<<(NM / 16) * (F_N / 16), 32, 0, stream>>>(c_it, m_it, c_emb, m_emb,
                                                      gat_w, XW);
    k_dots<<<(NM + D_N) / 8, 256, 0, stream>>>(c_it, m_it, c_emb, m_emb, hattr,
                                               VATT, WATT, S, T);
    k_edge<<<D_N, 256, 0, stream>>>(XW, S, T, EFEAT, A0);
    k_final<<<1, 256, 0, stream>>>(EFEAT, A0, gat_b, PE0S, PE1S, DIA, MED,
                                   lin_w, out);
}